// GNNBackbone_11450382811837
// MI455X (gfx1250) — compile-verified
//
#include <hip/hip_runtime.h>

#define N_NODES 50000
#define N_EDGES 800000
#define ET (N_EDGES + N_NODES)   // edges + self loops
#define NEG_SLOPE 0.2f
#define BN_EPS 1e-5f

typedef __attribute__((ext_vector_type(2))) float v2f;
typedef __attribute__((ext_vector_type(8))) float v8f;
typedef __attribute__((ext_vector_type(4))) unsigned int u32x4;
typedef __attribute__((ext_vector_type(8))) int i32x8;
typedef __attribute__((ext_vector_type(4))) int i32x4;

// ---- monotone float<->ordered-uint mapping for atomic max over floats ----
__device__ __forceinline__ unsigned f2ord(float f) {
  unsigned u = __float_as_uint(f);
  return (u & 0x80000000u) ? ~u : (u | 0x80000000u);
}
__device__ __forceinline__ float ord2f(unsigned k) {
  unsigned u = (k & 0x80000000u) ? (k & 0x7FFFFFFFu) : ~k;
  return __uint_as_float(u);
}

// ---------------------------------------------------------------------------
// GEMM: Hout[n x M] = X[n x K] * W[K x M], fp32 WMMA 16x16x4.
// W (<=64KB) is DMA'd once per workgroup into LDS by the Tensor Data Mover
// (tensor_load_to_lds + s_wait_tensorcnt), then each wave computes one
// 16-row strip across all M columns: the A fragment is loaded once per
// K-step and reused for all M/16 column tiles; B fragments come from LDS.
// Fragment layouts per CDNA5 ISA 7.12.2 (A: K0/K1 lanes 0-15, K2/K3 lanes
// 16-31; B/C/D row-striped across lanes, C/D VGPR v -> M=v / v+8).
// ---------------------------------------------------------------------------
template <int K, int M>
__global__ void gemm_wmma_tdm(const float* __restrict__ X,
                              const float* __restrict__ W,
                              float* __restrict__ Hout, int n) {
  constexpr int TM = M / 16;               // column tiles per row strip
  __shared__ float sW[K * M];              // W panel, row-major, LDS offset 0

  // --- TDM: DMA the whole K x M fp32 weight tensor into LDS (wave 0) ---
  if ((threadIdx.x >> 5) == 0) {
    unsigned long long ga = (unsigned long long)(uintptr_t)W;
    u32x4 g0;
    g0.x = 1u;                                         // count=1, user mode
    g0.y = 0u;                                         // lds_addr = 0 (sW base)
    g0.z = (unsigned)(ga & 0xFFFFFFFFu);               // global_addr[31:0]
    g0.w = (unsigned)((ga >> 32) & 0x1FFFFFFu)         // global_addr[56:32]
           | (2u << 30);                               // type = 2 ("image")
    i32x8 g1;
    g1.s0 = (int)(2u << 16);                           // data_size = 4 bytes
    g1.s1 = (int)((unsigned)M << 16);                  // tensor_dim0 = M
    g1.s2 = (int)((unsigned)K << 16);                  // tensor_dim1 = K
    g1.s3 = (int)((unsigned)M << 16);                  // tile_dim0 = M
    g1.s4 = (int)(unsigned)K;                          // tile_dim1 = K (tile_dim2=0)
    g1.s5 = (int)(unsigned)M;                          // tensor_dim0_stride = M
    g1.s6 = 0;
    g1.s7 = 0;
    i32x4 z4 = {0, 0, 0, 0};                           // dims 2..4 unused (2D tile)
    i32x8 z8 = {0, 0, 0, 0, 0, 0, 0, 0};
    __builtin_amdgcn_tensor_load_to_lds(g0, g1, z4, z4, z8, 0);
    __builtin_amdgcn_s_wait_tensorcnt(0);
  }
  __syncthreads();

  int wave = threadIdx.x >> 5;
  int lane = threadIdx.x & 31;
  int rowTile = blockIdx.x * (blockDim.x >> 5) + wave;
  if (rowTile >= (n >> 4)) return;
  int l15 = lane & 15;
  int koff = (lane >> 4) << 1;             // 0 for lanes 0-15, 2 for lanes 16-31
  const float* xrow = X + (size_t)(rowTile * 16 + l15) * K;

  v8f acc[TM];
#pragma unroll
  for (int t = 0; t < TM; ++t) acc[t] = (v8f){};

#pragma unroll 4
  for (int kk = 0; kk < K; kk += 4) {
    v2f a = *(const v2f*)(xrow + kk + koff);          // global_load_b64
    const float* bb = sW + (kk + koff) * M + l15;
#pragma unroll
    for (int t = 0; t < TM; ++t) {
      v2f b;
      b.x = bb[t * 16];                               // ds_load_b32
      b.y = bb[t * 16 + M];
      acc[t] = __builtin_amdgcn_wmma_f32_16x16x4_f32(false, a, false, b, 0,
                                                     acc[t], false, false);
    }
  }

  int rbase = rowTile * 16 + ((lane >> 4) << 3);      // +8 for hi half-wave
#pragma unroll
  for (int t = 0; t < TM; ++t) {
    int col = t * 16 + l15;
#pragma unroll
    for (int v = 0; v < 8; ++v)
      Hout[(size_t)(rbase + v) * M + col] = acc[t][v];
  }
}

// per-(node,head) attention logits: As = <h, a_src>, Ad = <h, a_dst>
__global__ void att_reduce(const float* __restrict__ Hbuf,
                           const float* __restrict__ a_src,
                           const float* __restrict__ a_dst,
                           float* __restrict__ As, float* __restrict__ Ad,
                           int n, int heads, int C) {
  int id = blockIdx.x * blockDim.x + threadIdx.x;
  if (id >= n * heads) return;
  int h = id % heads;
  const float* hp = Hbuf + (size_t)id * C;   // layout [n][H][C]
  const float* wsrc = a_src + h * C;
  const float* wdst = a_dst + h * C;
  float ss = 0.f, sd = 0.f;
  for (int c0 = 0; c0 < C; c0 += 4) {
    float4 hv = *(const float4*)(hp + c0);
    float4 sv = *(const float4*)(wsrc + c0);
    float4 dv = *(const float4*)(wdst + c0);
    ss += hv.x * sv.x + hv.y * sv.y + hv.z * sv.z + hv.w * sv.w;
    sd += hv.x * dv.x + hv.y * dv.y + hv.z * dv.z + hv.w * dv.w;
  }
  As[id] = ss;
  Ad[id] = sd;
}

__device__ __forceinline__ void edge_ends(const int* __restrict__ ei, int e,
                                          int& s, int& d) {
  if (e < N_EDGES) { s = ei[e]; d = ei[N_EDGES + e]; }
  else             { s = e - N_EDGES; d = s; }          // self loop
}

__device__ __forceinline__ float edge_logit(const float* __restrict__ As,
                                            const float* __restrict__ Ad,
                                            int s, int d, int heads, int h) {
  float ev = As[s * heads + h] + Ad[d * heads + h];
  return ev >= 0.f ? ev : NEG_SLOPE * ev;               // leaky relu
}

// pass 1: segment max per (dst, head), ordered-uint atomic max
__global__ void edge_max(const int* __restrict__ ei,
                         const float* __restrict__ As, const float* __restrict__ Ad,
                         unsigned* __restrict__ Smax, int heads) {
  int id = blockIdx.x * blockDim.x + threadIdx.x;
  if (id >= ET * heads) return;
  int e = id / heads, h = id - e * heads;
  int s, d; edge_ends(ei, e, s, d);
  float ev = edge_logit(As, Ad, s, d, heads, h);
  atomicMax(&Smax[d * heads + h], f2ord(ev));
}

// pass 2: segment sum of exp(e - max)
__global__ void edge_expsum(const int* __restrict__ ei,
                            const float* __restrict__ As, const float* __restrict__ Ad,
                            const unsigned* __restrict__ Smax,
                            float* __restrict__ Ssum, int heads) {
  int id = blockIdx.x * blockDim.x + threadIdx.x;
  if (id >= ET * heads) return;
  int e = id / heads, h = id - e * heads;
  int s, d; edge_ends(ei, e, s, d);
  float ev = edge_logit(As, Ad, s, d, heads, h);
  float ex = expf(ev - ord2f(Smax[d * heads + h]));
  atomicAdd(&Ssum[d * heads + h], ex);
}

// pass 3: Out[dst] += h[src] * alpha ; one thread per (edge, 4-channel group)
// -> b128 gathers from the L2-resident H buffer, alpha amortized over 4 ch.
__global__ void edge_agg(const int* __restrict__ ei,
                         const float* __restrict__ As, const float* __restrict__ Ad,
                         const unsigned* __restrict__ Smax,
                         const float* __restrict__ Ssum,
                         const float* __restrict__ Hbuf,
                         float* __restrict__ Out, int heads, int C) {
  int HC4 = (heads * C) >> 2;              // 4-channel groups per edge
  int id = blockIdx.x * blockDim.x + threadIdx.x;
  if (id >= ET * HC4) return;
  int e = id / HC4;
  int r4 = id - e * HC4;
  int c0 = r4 << 2;                        // base channel
  int h = c0 / C;
  int s, d; edge_ends(ei, e, s, d);
  float ev = edge_logit(As, Ad, s, d, heads, h);
  float alpha = expf(ev - ord2f(Smax[d * heads + h])) / Ssum[d * heads + h];
  int HC = heads * C;
  float4 hv = *(const float4*)(Hbuf + (size_t)s * HC + c0);   // global_load_b128
  float* ob = Out + (size_t)d * HC + c0;
  atomicAdd(ob + 0, hv.x * alpha);
  atomicAdd(ob + 1, hv.y * alpha);
  atomicAdd(ob + 2, hv.z * alpha);
  atomicAdd(ob + 3, hv.w * alpha);
}

// batch-norm stats: stats[c] = sum, stats[128+c] = sum of squares
__global__ void bn_stats(const float* __restrict__ X, float* __restrict__ stats,
                         int n, int CH) {
  __shared__ float s1[128];
  __shared__ float s2[128];
  for (int c = threadIdx.x; c < CH; c += blockDim.x) { s1[c] = 0.f; s2[c] = 0.f; }
  __syncthreads();
  int total = n * CH;
  for (int id = blockIdx.x * blockDim.x + threadIdx.x; id < total;
       id += gridDim.x * blockDim.x) {
    float v = X[id];
    int c = id % CH;
    atomicAdd(&s1[c], v);
    atomicAdd(&s2[c], v * v);
  }
  __syncthreads();
  for (int c = threadIdx.x; c < CH; c += blockDim.x) {
    atomicAdd(&stats[c], s1[c]);
    atomicAdd(&stats[128 + c], s2[c]);
  }
}

// y = (x - mu) * rsqrt(var + eps) * g + be ; optional relu
__global__ void bn_norm(const float* __restrict__ X, const float* __restrict__ stats,
                        const float* __restrict__ g, const float* __restrict__ be,
                        float* __restrict__ Y, int n, int CH, int relu) {
  int id = blockIdx.x * blockDim.x + threadIdx.x;
  if (id >= n * CH) return;
  int c = id % CH;
  float inv_n = 1.0f / (float)n;
  float mu = stats[c] * inv_n;
  float var = stats[128 + c] * inv_n - mu * mu;
  float y = (X[id] - mu) * rsqrtf(var + BN_EPS) * g[c] + be[c];
  if (relu) y = fmaxf(y, 0.f);
  Y[id] = y;
}

extern "C" void kernel_launch(void* const* d_in, const int* in_sizes, int n_in,
                              void* d_out, int out_size, void* d_ws, size_t ws_size,
                              hipStream_t stream) {
  const float* x   = (const float*)d_in[0];
  const int*   ei  = (const int*)d_in[1];
  const float* W0  = (const float*)d_in[2];
  const float* as0 = (const float*)d_in[3];
  const float* ad0 = (const float*)d_in[4];
  const float* g0  = (const float*)d_in[6];
  const float* be0 = (const float*)d_in[7];
  const float* W1  = (const float*)d_in[8];
  const float* as1 = (const float*)d_in[9];
  const float* ad1 = (const float*)d_in[10];
  const float* g1  = (const float*)d_in[12];
  const float* be1 = (const float*)d_in[13];
  const float* W2  = (const float*)d_in[14];
  const float* as2 = (const float*)d_in[15];
  const float* ad2 = (const float*)d_in[16];
  const float* g2  = (const float*)d_in[18];
  const float* be2 = (const float*)d_in[19];

  float* ws = (float*)d_ws;
  float* Hb = ws;                              // N*128 transformed features
  float* Xb = Hb + (size_t)N_NODES * 128;      // N*128 layer activations
  float* Ob = Xb + (size_t)N_NODES * 128;      // N*128 aggregation output
  float* As = Ob + (size_t)N_NODES * 128;      // N*4
  float* Ad = As + (size_t)N_NODES * 4;        // N*4
  unsigned* Sm = (unsigned*)(Ad + (size_t)N_NODES * 4);   // N*4 (ordered max)
  float* Ss = (float*)Sm + (size_t)N_NODES * 4;           // N*4
  float* St = Ss + (size_t)N_NODES * 4;                   // 256 floats stats

  // common (non-GEMM) part of one GAT layer
  auto run_tail = [&](int heads, int C, const float* aws, const float* awd,
                      const float* g, const float* be, float* out, int relu) {
    int M = heads * C;
    int nh = N_NODES * heads;
    att_reduce<<<(nh + 255) / 256, 256, 0, stream>>>(Hb, aws, awd, As, Ad,
                                                     N_NODES, heads, C);
    (void)hipMemsetAsync(Sm, 0, (size_t)N_NODES * heads * 4, stream);
    (void)hipMemsetAsync(Ss, 0, (size_t)N_NODES * heads * 4, stream);
    (void)hipMemsetAsync(Ob, 0, (size_t)N_NODES * M * 4, stream);
    (void)hipMemsetAsync(St, 0, 256 * 4, stream);
    int eh = ET * heads;
    edge_max<<<(eh + 255) / 256, 256, 0, stream>>>(ei, As, Ad, Sm, heads);
    edge_expsum<<<(eh + 255) / 256, 256, 0, stream>>>(ei, As, Ad, Sm, Ss, heads);
    int ec = ET * (M >> 2);
    edge_agg<<<(ec + 255) / 256, 256, 0, stream>>>(ei, As, Ad, Sm, Ss, Hb, Ob,
                                                   heads, C);
    bn_stats<<<1024, 256, 0, stream>>>(Ob, St, N_NODES, M);
    bn_norm<<<((N_NODES * M) + 255) / 256, 256, 0, stream>>>(Ob, St, g, be, out,
                                                             N_NODES, M, relu);
  };

  int rowTiles = N_NODES / 16;                 // 3125
  int gemmBlocks = (rowTiles + 7) / 8;         // 8 waves (row strips) per block

  // layer 0: 32 -> 4x32 concat, BN+ReLU
  gemm_wmma_tdm<32, 128><<<gemmBlocks, 256, 0, stream>>>(x, W0, Hb, N_NODES);
  run_tail(4, 32, as0, ad0, g0, be0, Xb, 1);
  // layer 1: 128 -> 4x32 concat, BN+ReLU
  gemm_wmma_tdm<128, 128><<<gemmBlocks, 256, 0, stream>>>(Xb, W1, Hb, N_NODES);
  run_tail(4, 32, as1, ad1, g1, be1, Xb, 1);
  // layer 2: 128 -> 1x32 (mean over 1 head == identity), BN, no ReLU
  gemm_wmma_tdm<128, 32><<<gemmBlocks, 256, 0, stream>>>(Xb, W2, Hb, N_NODES);
  run_tail(1, 32, as2, ad2, g2, be2, (float*)d_out, 0);
}